// MSDeformAttnFix_214748365281
// MI455X (gfx1250) — compile-verified
//
#include <hip/hip_runtime.h>
#include <hip/hip_bf16.h>
#include <cstddef>

// ---------------------------------------------------------------------------
// MS-Deformable-Attention-3D forward for MI455X (gfx1250, wave32, WMMA).
//   1) offs  = query @ w_off^T + b_off            (WMMA f32 16x16x4 GEMM)
//   2) qproj = query @ wq^T + bq                  (same GEMM kernel)
//   3) trilinear sampling -> scrambled kv buffer  (faithful raw-view replay)
//   4) fused per-token attention: async-LDS kv stage, k/v-proj (WMMA),
//      half-wave shfl softmax, weighted sum
//   5) oproj = attn_out @ out_proj_w^T + b        (same GEMM kernel)
//   6) faithful (bs,256,Lq)->(bs,Lq,256) transpose into d_out
// ---------------------------------------------------------------------------

typedef __attribute__((ext_vector_type(2))) float v2f;
typedef __attribute__((ext_vector_type(8))) float v8f;
typedef __attribute__((ext_vector_type(4))) int   v4i;

// typed address-space pointers for the async-to-LDS builtin
typedef __attribute__((address_space(1))) v4i gas_v4i;
typedef __attribute__((address_space(3))) v4i las_v4i;

#if defined(__has_builtin)
#if __has_builtin(__builtin_amdgcn_wmma_f32_16x16x4_f32)
#define USE_WMMA_F32X4 1
#endif
#if __has_builtin(__builtin_amdgcn_global_load_async_to_lds_b128) && \
    __has_builtin(__builtin_amdgcn_s_wait_asynccnt)
#define USE_ASYNC_LDS 1
#endif
#endif

#define BS 2
#define LQ 8192
#define DM 256
#define NH 8
#define NL 4
#define NP 4
#define CH 32
#define LEN_IN 126360

// ws layout (in floats)
#define OFFS_OFF   0LL
#define QPROJ_OFF  (OFFS_OFF  + 16384LL * 384)
#define ATTN_OFF   (QPROJ_OFF + 16384LL * 256)
#define OPROJ_OFF  (ATTN_OFF  + 16384LL * 256)
#define KV_OFF     (OPROJ_OFF + 16384LL * 256)

// ---------------------------------------------------------------------------
// C(M,N) = X(M,K) @ W(N,K)^T + bias(N), f32, WMMA 16x16x4.
// One wave computes a 16x64 strip: one shared A fragment, 4 N-tiles /
// 4 accumulators. Register double-buffered: next k-step's 5 fragments are
// loaded before the current step's 4 WMMAs consume theirs.
// M mult of 16, N mult of 64, K mult of 4, exact grid. 8 waves / block.
// ---------------------------------------------------------------------------
__global__ void gemm_bias_f32_wmma(const float* __restrict__ X,
                                   const float* __restrict__ W,
                                   const float* __restrict__ bias,
                                   float* __restrict__ C,
                                   int M, int N, int K) {
  const int lane = threadIdx.x & 31;
  const int wv   = threadIdx.x >> 5;
  const int ngrp = N >> 6;                 // groups of 4 n-tiles
  const int gid  = blockIdx.x * 8 + wv;
  const int m0 = (gid / ngrp) << 4;
  const int n0 = (gid % ngrp) << 6;
  const int lr = lane & 15;
  const int hi = lane >> 4;

  const float* xrow = X + (size_t)(m0 + lr) * K + 2 * hi;
  const float* wrow0 = W + (size_t)(n0 +  0 + lr) * K + 2 * hi;
  const float* wrow1 = W + (size_t)(n0 + 16 + lr) * K + 2 * hi;
  const float* wrow2 = W + (size_t)(n0 + 32 + lr) * K + 2 * hi;
  const float* wrow3 = W + (size_t)(n0 + 48 + lr) * K + 2 * hi;

  v8f acc0 = {0.f,0.f,0.f,0.f,0.f,0.f,0.f,0.f};
  v8f acc1 = acc0, acc2 = acc0, acc3 = acc0;

#ifdef USE_WMMA_F32X4
  v2f a  = *(const v2f*)(xrow);
  v2f b0 = *(const v2f*)(wrow0);
  v2f b1 = *(const v2f*)(wrow1);
  v2f b2 = *(const v2f*)(wrow2);
  v2f b3 = *(const v2f*)(wrow3);
  #pragma unroll 4
  for (int k = 4; k < K; k += 4) {
    // prefetch next k-step while current WMMAs run
    v2f an  = *(const v2f*)(xrow + k);
    v2f b0n = *(const v2f*)(wrow0 + k);
    v2f b1n = *(const v2f*)(wrow1 + k);
    v2f b2n = *(const v2f*)(wrow2 + k);
    v2f b3n = *(const v2f*)(wrow3 + k);
    acc0 = __builtin_amdgcn_wmma_f32_16x16x4_f32(false, a, false, b0, (short)0, acc0, false, false);
    acc1 = __builtin_amdgcn_wmma_f32_16x16x4_f32(false, a, false, b1, (short)0, acc1, false, false);
    acc2 = __builtin_amdgcn_wmma_f32_16x16x4_f32(false, a, false, b2, (short)0, acc2, false, false);
    acc3 = __builtin_amdgcn_wmma_f32_16x16x4_f32(false, a, false, b3, (short)0, acc3, false, false);
    a = an; b0 = b0n; b1 = b1n; b2 = b2n; b3 = b3n;
  }
  acc0 = __builtin_amdgcn_wmma_f32_16x16x4_f32(false, a, false, b0, (short)0, acc0, false, false);
  acc1 = __builtin_amdgcn_wmma_f32_16x16x4_f32(false, a, false, b1, (short)0, acc1, false, false);
  acc2 = __builtin_amdgcn_wmma_f32_16x16x4_f32(false, a, false, b2, (short)0, acc2, false, false);
  acc3 = __builtin_amdgcn_wmma_f32_16x16x4_f32(false, a, false, b3, (short)0, acc3, false, false);
#else
  {
    v8f* accs[4] = {&acc0, &acc1, &acc2, &acc3};
    for (int t = 0; t < 4; ++t) {
      const float* wfull = W + (size_t)(n0 + 16 * t + lr) * K;
      #pragma unroll
      for (int r = 0; r < 8; ++r) {
        const float* xfull = X + (size_t)(m0 + hi * 8 + r) * K;
        float s = 0.f;
        for (int k = 0; k < K; ++k) s += xfull[k] * wfull[k];
        (*accs[t])[r] = s;
      }
    }
  }
#endif

  {
    v8f accs[4] = {acc0, acc1, acc2, acc3};
    #pragma unroll
    for (int t = 0; t < 4; ++t) {
      const int col = n0 + 16 * t + lr;
      const float bb = bias[col];
      float* crow = C + (size_t)(m0 + hi * 8) * N + col;
      #pragma unroll
      for (int r = 0; r < 8; ++r) crow[(size_t)r * N] = accs[t][r] + bb;
    }
  }
}

// ---------------------------------------------------------------------------
// Trilinear sampling with faithful scrambled scatter into kv.
// One wave per sample point (b,q,h,l,p); lane = channel (0..31).
// ---------------------------------------------------------------------------
__global__ void sample_kernel(const float* __restrict__ offs,
                              const float* __restrict__ refp,
                              const float* __restrict__ inp,
                              float* __restrict__ kv) {
  const int LSTART[4] = {0, 110592, 124416, 126144};
  const int LSZ[4]    = {48, 24, 12, 6};

  const int lane = threadIdx.x & 31;
  const int pid  = blockIdx.x * 8 + (threadIdx.x >> 5);
  const int p = pid & 3;
  const int l = (pid >> 2) & 3;
  const int h = (pid >> 4) & 7;
  const int q = (pid >> 7) & 8191;
  const int b = pid >> 20;

  const int s = LSZ[l];
  const float fs = (float)s;

  const size_t bq = (size_t)(b * LQ + q);
  const size_t obase = bq * 384 + (size_t)(((h * 4 + l) * 4 + p) * 3);
  const size_t rbase = (bq * 4 + l) * 3;

  const float o0 = offs[obase + 0], o1 = offs[obase + 1], o2 = offs[obase + 2];
  const float r0 = refp[rbase + 0], r1 = refp[rbase + 1], r2 = refp[rbase + 2];

  // loc = ref + off/norm ; grid = relu(2*loc - 1)
  const float g0 = fmaxf(0.f, 2.f * (r0 + o0 / fs) - 1.f);  // x -> W
  const float g1 = fmaxf(0.f, 2.f * (r1 + o1 / fs) - 1.f);  // y -> H
  const float g2 = fmaxf(0.f, 2.f * (r2 + o2 / fs) - 1.f);  // z -> D

  const float ix = ((g0 + 1.f) * fs - 1.f) * 0.5f;
  const float iy = ((g1 + 1.f) * fs - 1.f) * 0.5f;
  const float iz = ((g2 + 1.f) * fs - 1.f) * 0.5f;

  const float x0f = floorf(ix), y0f = floorf(iy), z0f = floorf(iz);
  const int x0 = (int)x0f, y0 = (int)y0f, z0 = (int)z0f;
  const float fx = ix - x0f, fy = iy - y0f, fz = iz - z0f;

  const float* src = inp + ((size_t)b * LEN_IN + LSTART[l]) * 256 + h * CH + lane;

  float acc = 0.f;
  #pragma unroll
  for (int dz = 0; dz < 2; ++dz) {
    const int zi = z0 + dz;
    const bool vz = (zi >= 0) && (zi < s);
    const float wz = dz ? fz : 1.f - fz;
    #pragma unroll
    for (int dy = 0; dy < 2; ++dy) {
      const int yi = y0 + dy;
      const bool vy = (yi >= 0) && (yi < s);
      const float wy = dy ? fy : 1.f - fy;
      #pragma unroll
      for (int dx = 0; dx < 2; ++dx) {
        const int xi = x0 + dx;
        const bool vx = (xi >= 0) && (xi < s);
        const float wx = dx ? fx : 1.f - fx;
        if (vz && vy && vx) {
          const int lin = (zi * s + yi) * s + xi;
          acc += (wz * wy * wx) * src[(size_t)lin * 256];
        }
      }
    }
  }

  // faithful raw view: s_l (bs*H, c, Lq*P) -> (bs*Lq, P, H*c), concat on axis1
  const int outer = (b * NH + h) * CH + lane;
  const int qp4   = q * NP + p;
  const int i     = outer * 32 + (qp4 >> 10);
  const int inner = qp4 & 1023;
  const int pp    = inner >> 8;
  const int j     = inner & 255;
  kv[(size_t)i * 4096 + (size_t)((l * 4 + pp) * 256 + j)] = acc;
}

// ---------------------------------------------------------------------------
// Fused per-token attention. One block (512 threads / 16 waves) per token i.
//  - stage kv[i] (16x256) in LDS (async-to-LDS b128 when available)
//  - 16 waves: WMMA k-proj + v-proj 16x16 tiles (shared A fragment,
//    register double-buffered)
//  - 128 lanes: logits + half-wave shfl softmax (16 keys per head)
//  - 256 lanes: o[j] = sum_k p[h][k] * vproj[k][j]
// ---------------------------------------------------------------------------
#define KVS_STRIDE 260   // mult of 4 (16B-aligned b128 rows), bank stride 4
#define PRJ_STRIDE 257

__global__ void attn_kernel(const float* __restrict__ kv,
                            const float* __restrict__ qp,
                            const float* __restrict__ wK,
                            const float* __restrict__ wV,
                            const float* __restrict__ bK,
                            const float* __restrict__ bV,
                            float* __restrict__ out) {
  __shared__ float kvs[16 * KVS_STRIDE];
  __shared__ float kps[16 * PRJ_STRIDE];
  __shared__ float vps[16 * PRJ_STRIDE];
  __shared__ float probs[NH][16];

  const int tid = threadIdx.x;
  const int i = blockIdx.x;

  // phase 1: stage kv row-block (16x256 floats)
  const float* kvrow = kv + (size_t)i * 4096;
#ifdef USE_ASYNC_LDS
  #pragma unroll
  for (int c4 = tid; c4 < 1024; c4 += 512) {      // 1024 x 16B chunks
    const int row = c4 >> 6;
    const int col = (c4 & 63) << 2;
    __builtin_amdgcn_global_load_async_to_lds_b128(
        (gas_v4i*)(kvrow + (size_t)c4 * 4),
        (las_v4i*)&kvs[row * KVS_STRIDE + col],
        0, 0);
  }
  __builtin_amdgcn_s_wait_asynccnt(0);
#else
  #pragma unroll
  for (int e = tid; e < 4096; e += 512) {
    kvs[(e >> 8) * KVS_STRIDE + (e & 255)] = kvrow[e];
  }
#endif
  __syncthreads();

  // phase 2: k/v projections (16 keys x 256 ch), wave w -> cols [16w,16w+16)
  const int lane = tid & 31;
  const int wv   = tid >> 5;
  const int lr = lane & 15;
  const int hi = lane >> 4;
  const int n0 = wv << 4;

  const float* arow  = &kvs[lr * KVS_STRIDE + 2 * hi];
  const float* wkrow = wK + (size_t)(n0 + lr) * 256 + 2 * hi;
  const float* wvrow = wV + (size_t)(n0 + lr) * 256 + 2 * hi;

  v8f ak = {0.f,0.f,0.f,0.f,0.f,0.f,0.f,0.f};
  v8f av = ak;
#ifdef USE_WMMA_F32X4
  v2f a  = *(const v2f*)(arow);
  v2f b0 = *(const v2f*)(wkrow);
  v2f b1 = *(const v2f*)(wvrow);
  #pragma unroll 4
  for (int k = 4; k < 256; k += 4) {
    v2f an  = *(const v2f*)(arow + k);
    v2f b0n = *(const v2f*)(wkrow + k);
    v2f b1n = *(const v2f*)(wvrow + k);
    ak = __builtin_amdgcn_wmma_f32_16x16x4_f32(false, a, false, b0, (short)0, ak, false, false);
    av = __builtin_amdgcn_wmma_f32_16x16x4_f32(false, a, false, b1, (short)0, av, false, false);
    a = an; b0 = b0n; b1 = b1n;
  }
  ak = __builtin_amdgcn_wmma_f32_16x16x4_f32(false, a, false, b0, (short)0, ak, false, false);
  av = __builtin_amdgcn_wmma_f32_16x16x4_f32(false, a, false, b1, (short)0, av, false, false);
#else
  {
    const float* wk_full = wK + (size_t)(n0 + lr) * 256;
    const float* wv_full = wV + (size_t)(n0 + lr) * 256;
    #pragma unroll
    for (int r = 0; r < 8; ++r) {
      const float* af = &kvs[(hi * 8 + r) * KVS_STRIDE];
      float sk = 0.f, sv = 0.f;
      for (int k = 0; k < 256; ++k) { sk += af[k] * wk_full[k]; sv += af[k] * wv_full[k]; }
      ak[r] = sk; av[r] = sv;
    }
  }
#endif
  {
    const int ch = n0 + lr;
    const float bbk = bK[ch];
    const float bbv = bV[ch];
    #pragma unroll
    for (int r = 0; r < 8; ++r) {
      const int key = hi * 8 + r;
      kps[key * PRJ_STRIDE + ch] = ak[r] + bbk;
      vps[key * PRJ_STRIDE + ch] = av[r] + bbv;
    }
  }
  __syncthreads();

  // phase 3: logits + softmax (128 lanes: h = tid>>4, k = tid&15)
  if (tid < 128) {
    const int h = tid >> 4;
    const int k = tid & 15;
    const float* qrow = qp + (size_t)i * 256 + h * CH;
    const float* krow = &kps[k * PRJ_STRIDE + h * CH];
    float lg = 0.f;
    #pragma unroll
    for (int c = 0; c < CH; ++c) lg += qrow[c] * krow[c];
    lg *= 0.17677669529663687f;  // 1/sqrt(32)

    float mx = lg;
    #pragma unroll
    for (int m = 8; m >= 1; m >>= 1) mx = fmaxf(mx, __shfl_xor(mx, m, 16));
    float ev = expf(lg - mx);
    float sm = ev;
    #pragma unroll
    for (int m = 8; m >= 1; m >>= 1) sm += __shfl_xor(sm, m, 16);
    probs[h][k] = ev / sm;
  }
  __syncthreads();

  // phase 4: o[j] = sum_k p[h][k] * vproj[k][j]
  if (tid < 256) {
    const int j = tid;
    const int h = j >> 5;
    float acc = 0.f;
    #pragma unroll
    for (int k = 0; k < 16; ++k) acc += probs[h][k] * vps[k * PRJ_STRIDE + j];
    out[(size_t)i * 256 + j] = acc;
  }
}

// ---------------------------------------------------------------------------
// Faithful final view: out[b,q,d] = oproj_flat[(b*256+d)*8192 + q]
// i.e. a (256 x 8192) transpose per batch, via LDS tiles.
// ---------------------------------------------------------------------------
__global__ void final_transpose(const float* __restrict__ src,
                                float* __restrict__ dst) {
  __shared__ float tile[32][33];
  const int b  = blockIdx.z;
  const int q0 = blockIdx.x * 32;
  const int d0 = blockIdx.y * 32;
  const int tx = threadIdx.x;  // 0..31
  const int ty = threadIdx.y;  // 0..7
  #pragma unroll
  for (int r = 0; r < 4; ++r) {
    const int d = d0 + ty + 8 * r;
    tile[ty + 8 * r][tx] = src[((size_t)b * 256 + d) * 8192 + q0 + tx];
  }
  __syncthreads();
  #pragma unroll
  for (int r = 0; r < 4; ++r) {
    const int q = q0 + ty + 8 * r;
    dst[((size_t)b * LQ + q) * 256 + d0 + tx] = tile[tx][ty + 8 * r];
  }
}

// ---------------------------------------------------------------------------
extern "C" void kernel_launch(void* const* d_in, const int* in_sizes, int n_in,
                              void* d_out, int out_size, void* d_ws, size_t ws_size,
                              hipStream_t stream) {
  const float* query   = (const float*)d_in[0];  // (2,8192,256)
  const float* refp    = (const float*)d_in[1];  // (2,8192,4,3)
  const float* inp     = (const float*)d_in[2];  // (2,126360,256)
  const float* w_off   = (const float*)d_in[3];  // (384,256)
  const float* b_off   = (const float*)d_in[4];  // (384)
  const float* in_w    = (const float*)d_in[5];  // (768,256)
  const float* in_b    = (const float*)d_in[6];  // (768)
  const float* out_w   = (const float*)d_in[7];  // (256,256)
  const float* out_b   = (const float*)d_in[8];  // (256)
  // d_in[9], d_in[10]: spatial shapes / level starts -> static, hardcoded.

  float* ws    = (float*)d_ws;
  float* offs  = ws + OFFS_OFF;
  float* qproj = ws + QPROJ_OFF;
  float* attn  = ws + ATTN_OFF;
  float* oproj = ws + OPROJ_OFF;
  float* kvbuf = ws + KV_OFF;
  float* outp  = (float*)d_out;

  // 1) sampling offsets: (16384,384) = query @ w_off^T + b_off
  gemm_bias_f32_wmma<<<(16384 / 16) * (384 / 64) / 8, 256, 0, stream>>>(
      query, w_off, b_off, offs, 16384, 384, 256);

  // 2) q projection: (16384,256) = query @ wq^T + bq
  gemm_bias_f32_wmma<<<(16384 / 16) * (256 / 64) / 8, 256, 0, stream>>>(
      query, in_w, in_b, qproj, 16384, 256, 256);

  // 3) trilinear sampling -> scrambled kv (2,097,152 points, 8 per block)
  sample_kernel<<<2097152 / 8, 256, 0, stream>>>(offs, refp, inp, kvbuf);

  // 4) fused attention per token
  attn_kernel<<<16384, 512, 0, stream>>>(kvbuf, qproj,
                                         in_w + 256 * 256, in_w + 512 * 256,
                                         in_b + 256, in_b + 512, attn);

  // 5) out projection
  gemm_bias_f32_wmma<<<(16384 / 16) * (256 / 64) / 8, 256, 0, stream>>>(
      attn, out_w, out_b, oproj, 16384, 256, 256);

  // 6) faithful final transpose into d_out
  final_transpose<<<dim3(8192 / 32, 256 / 32, 2), dim3(32, 8), 0, stream>>>(
      oproj, outp);
}